// Gate_32375463478041
// MI455X (gfx1250) — compile-verified
//
#include <hip/hip_runtime.h>
#include <math.h>

typedef float v2f __attribute__((ext_vector_type(2)));
typedef float v4f __attribute__((ext_vector_type(4)));
typedef float v8f __attribute__((ext_vector_type(8)));

#define N_BATCH   32
#define N_SEQ     4096
#define N_CH      128
#define N_EXP     64
#define D_IN      (2 * N_SEQ)      // 8192 features per row
#define NK4       (D_IN / 4)       // 2048 K-steps of 4 (one WMMA each)
#define NT        8                // fused N tiles: 4 (main) + 4 (noise), N=128
#define WAVES     8
#define N_ROWS    (N_BATCH * N_CH) // 4096 rows

// ---------------------------------------------------------------------------
// Kernel 0: repack W_main / W_noise into per-lane-contiguous B-fragment layout:
//   pk[ ((k*32 + lane)*NT + t)*2 + w ]
//     t<4 : W_main [e = 16t     + lane%16, f = 4k + 2*(lane>=16) + w]
//     t>=4: W_noise[e = 16(t-4) + lane%16, same f]
// Each lane's 8 fragments for K-step k are 64 contiguous bytes -> the GEMM
// reads them with 4 global_load_b128 (one contiguous 2KB block per wave).
// ---------------------------------------------------------------------------
__global__ void Gate_repack_kernel(const float* __restrict__ Wm,
                                   const float* __restrict__ Wn,
                                   float* __restrict__ pk) {
    int idx = blockIdx.x * blockDim.x + threadIdx.x;   // (k,lane,t) flat
    if (idx >= NK4 * 32 * NT) return;
    int t    = idx & 7;
    int lane = (idx >> 3) & 31;
    int k    = idx >> 8;
    const float* W = (t < 4) ? Wm : Wn;
    int e  = ((t & 3) << 4) + (lane & 15);
    int f0 = 4 * k + 2 * (lane >> 4);
    size_t src = (size_t)e * D_IN + f0;
    pk[(size_t)idx * 2 + 0] = W[src + 0];
    pk[(size_t)idx * 2 + 1] = W[src + 1];
}

// ---------------------------------------------------------------------------
// Kernel 1: fused dual GEMM, software-pipelined with ping-pong buffers.
//   logits[row, 0:64] = x_r @ Wm.T ; logits[row, 64:128] = x_r @ Wn.T
// grid: 32 batches x 8 channel-tiles, block: 256 (8 wave32), wave w owns
// K-steps k = w, w+8, ... Loop consumes buffer A while loading buffer B
// (and vice versa) -> no rotation copies, loads a full pair ahead of use.
// ---------------------------------------------------------------------------
template <int T>
__device__ __forceinline__ void wmma_tile(v8f acc[NT], v2f a, const v4f bb[4]) {
    v2f bf = __builtin_shufflevector(bb[T >> 1], bb[T >> 1],
                                     (T & 1) * 2, (T & 1) * 2 + 1);
    acc[T] = __builtin_amdgcn_wmma_f32_16x16x4_f32(
        false, a, false, bf, (short)0, acc[T], false, false);
}

__device__ __forceinline__ void wmma8(v8f acc[NT], v2f a, const v4f bb[4]) {
    wmma_tile<0>(acc, a, bb);
    wmma_tile<1>(acc, a, bb);
    wmma_tile<2>(acc, a, bb);
    wmma_tile<3>(acc, a, bb);
    wmma_tile<4>(acc, a, bb);
    wmma_tile<5>(acc, a, bb);
    wmma_tile<6>(acc, a, bb);
    wmma_tile<7>(acc, a, bb);
}

__global__ __launch_bounds__(256) void Gate_gemm_kernel(
        const float* __restrict__ x,
        const float* __restrict__ pk,
        float* __restrict__ logits) {
    const int b    = blockIdx.x >> 3;     // batch
    const int ct   = blockIdx.x & 7;      // channel tile (16 channels)
    const int lane = threadIdx.x & 31;
    const int w    = threadIdx.x >> 5;    // wave id 0..7
    const int lhi  = lane >> 4;           // lane half (K pair select)
    const int llo  = lane & 15;

    v8f acc[NT] = {};                     // 8 tiles x 8 VGPR f32 accumulators

    // A-frag (16x4 f32): lane l<16 -> M=l,K=0/1 ; lane l+16 -> M=l,K=2/3.
    // float2 index into x: ((b*4096 + s)*128 + c), s = 2k + lhi, c = ct*16+llo
    const v2f* __restrict__ Abase =
        (const v2f*)x + ((size_t)b * N_SEQ + lhi) * N_CH + ct * 16 + llo;
    // B: 4 x v4f per lane per K-step, contiguous 64B
    const v4f* __restrict__ Bbase = (const v4f*)pk + (size_t)lane * 4;

    v2f a0, a1;
    v4f b0[4], b1[4];

    // prologue: load the first pair of K-steps (w, w+8)
    a0 = Abase[(size_t)(2 * w) * N_CH];
    {
        const v4f* Bp = Bbase + (size_t)w * 128;
#pragma unroll
        for (int j = 0; j < 4; ++j) b0[j] = Bp[j];
    }
    a1 = Abase[(size_t)(2 * (w + WAVES)) * N_CH];
    {
        const v4f* Bp = Bbase + (size_t)(w + WAVES) * 128;
#pragma unroll
        for (int j = 0; j < 4; ++j) b1[j] = Bp[j];
    }

    // steady state: consume pair (k, k+8), preload pair (k+16, k+24)
    int k = w;
    for (; k + 3 * WAVES < NK4; k += 2 * WAVES) {
        const int k2 = k + 2 * WAVES;
        const int k3 = k + 3 * WAVES;

        wmma8(acc, a0, b0);               // consume step k
        a0 = Abase[(size_t)(2 * k2) * N_CH];
        __builtin_prefetch(&Abase[(size_t)(2 * (k2 + 2 * WAVES)) * N_CH], 0, 3);
        {
            const v4f* Bp = Bbase + (size_t)k2 * 128;
#pragma unroll
            for (int j = 0; j < 4; ++j) b0[j] = Bp[j];
        }

        wmma8(acc, a1, b1);               // consume step k+8
        a1 = Abase[(size_t)(2 * k3) * N_CH];
        {
            const v4f* Bp = Bbase + (size_t)k3 * 128;
#pragma unroll
            for (int j = 0; j < 4; ++j) b1[j] = Bp[j];
        }
    }
    wmma8(acc, a0, b0);                   // final pair (w+2032, w+2040)
    wmma8(acc, a1, b1);

    // Cross-wave K reduction: one tile per round, wave t finalizes tile t.
    __shared__ v8f red[WAVES][32];        // 8 KB
    const int row0 = b * N_CH + ct * 16;
#pragma unroll
    for (int t = 0; t < NT; ++t) {
        red[w][lane] = acc[t];
        __syncthreads();
        if (w == t) {
            v8f s = red[0][lane];
#pragma unroll
            for (int j = 1; j < WAVES; ++j) s += red[j][lane];
            // C layout: lane llo = N, lane half -> M+8, vgpr v = M row
#pragma unroll
            for (int v = 0; v < 8; ++v) {
                int M = v + 8 * lhi;
                logits[(size_t)(row0 + M) * (2 * N_EXP) + t * 16 + llo] = s[v];
            }
        }
        __syncthreads();
    }
}

// ---------------------------------------------------------------------------
// Kernel 2: per-row gating epilogue.
// g = (main + b_main) + softplus(noise + b_noise); keep strictly greater than
// the 3rd-largest (kthvalue(n-k), duplicates counted), softmax over kept.
// ---------------------------------------------------------------------------
__device__ __forceinline__ float gate_val(const float* Lr,
                                          const float* bm, const float* bn,
                                          int e) {
    float m = Lr[e] + bm[e];
    float z = Lr[N_EXP + e] + bn[e];
    float sp = fmaxf(z, 0.0f) + log1pf(expf(-fabsf(z)));  // stable softplus
    return m + sp;
}

__global__ void Gate_softmax_kernel(const float* __restrict__ logits,
                                    const float* __restrict__ bm,
                                    const float* __restrict__ bn,
                                    float* __restrict__ out) {
    int row = blockIdx.x * blockDim.x + threadIdx.x;
    if (row >= N_ROWS) return;
    const float* Lr = logits + (size_t)row * (2 * N_EXP);

    float t1 = -INFINITY, t2 = -INFINITY, t3 = -INFINITY;  // multiset top-3
    for (int e = 0; e < N_EXP; ++e) {
        float g = gate_val(Lr, bm, bn, e);
        if (g > t1)      { t3 = t2; t2 = t1; t1 = g; }
        else if (g > t2) { t3 = t2; t2 = g; }
        else if (g > t3) { t3 = g; }
    }
    const float thr = t3;        // kth = sorted[n-k-1] (3rd largest), strict >
    const float mx  = t1;
    float sum = 0.0f;
    for (int e = 0; e < N_EXP; ++e) {
        float g = gate_val(Lr, bm, bn, e);
        if (g > thr) sum += expf(g - mx);
    }
    float inv = 1.0f / sum;
    for (int e = 0; e < N_EXP; ++e) {
        float g = gate_val(Lr, bm, bn, e);
        out[(size_t)row * N_EXP + e] = (g > thr) ? expf(g - mx) * inv : 0.0f;
    }
}

// ---------------------------------------------------------------------------
extern "C" void kernel_launch(void* const* d_in, const int* in_sizes, int n_in,
                              void* d_out, int out_size, void* d_ws, size_t ws_size,
                              hipStream_t stream) {
    const float* x  = (const float*)d_in[0];   // [32,4096,128,2]
    const float* Wm = (const float*)d_in[1];   // [64,8192]
    const float* bm = (const float*)d_in[2];   // [64]
    const float* Wn = (const float*)d_in[3];   // [64,8192]
    const float* bn = (const float*)d_in[4];   // [64]
    float* out = (float*)d_out;                // [4096,64]

    float* pk     = (float*)d_ws;                          // 4 MB packed W
    float* logits = pk + (size_t)NK4 * 32 * NT * 2;        // 2 MB raw logits

    // 0) repack weights into per-lane-contiguous WMMA B-fragment order
    {
        int total = NK4 * 32 * NT;                         // 524288
        Gate_repack_kernel<<<(total + 255) / 256, 256, 0, stream>>>(Wm, Wn, pk);
    }
    // 1) fused dual GEMM with f32 WMMA (ping-pong software pipeline)
    Gate_gemm_kernel<<<N_BATCH * 8, 256, 0, stream>>>(x, pk, logits);
    // 2) bias + softplus + top-2 mask + softmax
    Gate_softmax_kernel<<<(N_ROWS + 255) / 256, 256, 0, stream>>>(logits, bm, bn, out);
}